// KeypointLoss_15023795601463
// MI455X (gfx1250) — compile-verified
//
#include <hip/hip_runtime.h>
#include <math.h>

typedef __attribute__((ext_vector_type(2))) float v2f;
typedef __attribute__((ext_vector_type(8))) float v8f;

#define KEYS 11
#define NSTACK 4
#define BATCH 16
#define HH 128
#define WW 128
#define MAP (HH * WW)          // 16384 elements per (b,s,k) map
#define NMAPS (BATCH * NSTACK * KEYS)  // 704

// ---------------------------------------------------------------------------
// Stage 1: one block per (b,s,k) heatmap (704 blocks, 256 threads = 8 waves).
// Streams 16384 floats of hm + heatmap with b128 loads, accumulates
// (hm-h)^2 into a WMMA f32 16x16x4 accumulator (B = ones => row-sum engine),
// and tracks max/argmax (first-occurrence semantics) per lane.
// ---------------------------------------------------------------------------
__global__ __launch_bounds__(256) void kploss_stage1(
    const float* __restrict__ cp,      // (B, NSTACK, 2*KEYS, H, W)
    const float* __restrict__ hmaps,   // (B, KEYS, H, W)
    float* __restrict__ ws_sum,        // [704]
    float* __restrict__ ws_max,        // [704]
    int*   __restrict__ ws_idx)        // [704]
{
    const int blk = blockIdx.x;            // b*NSTACK*KEYS + s*KEYS + k
    const int k   = blk % KEYS;
    const int bs  = blk / KEYS;            // b*NSTACK + s
    const int b   = bs / NSTACK;

    const float* __restrict__ src  = cp    + (size_t)(bs * (2 * KEYS) + k) * MAP;
    const float* __restrict__ hsrc = hmaps + (size_t)(b * KEYS + k) * MAP;

    const int tid = threadIdx.x;
    const int w   = tid >> 5;              // wave id 0..7
    const int L   = tid & 31;              // lane

    v8f acc = {};                          // WMMA C/D accumulator (f32 16x16)
    v2f onesv; onesv[0] = 1.0f; onesv[1] = 1.0f;   // B = ones (4x16)

    float bestv = -INFINITY;
    int   besti = 0;

    // 16 iterations: each wave eats 128 contiguous floats (float4 per lane).
    for (int t = 0; t < 16; ++t) {
        const int off = t * 1024 + w * 128 + L * 4;
        const float4 x = *(const float4*)(src + off);
        const float4 h = *(const float4*)(hsrc + off);
        if (t < 15) {
            __builtin_prefetch(src + off + 1024, 0, 1);   // global_prefetch_b8
            __builtin_prefetch(hsrc + off + 1024, 0, 1);
        }

        const float d0 = x.x - h.x, d1 = x.y - h.y;
        const float d2 = x.z - h.z, d3 = x.w - h.w;
        v2f a0; a0[0] = d0 * d0; a0[1] = d1 * d1;
        v2f a1; a1[0] = d2 * d2; a1[1] = d3 * d3;

        // D = A*ones + C : accumulates sum of all A elements into row-sums.
        acc = __builtin_amdgcn_wmma_f32_16x16x4_f32(
                  false, a0, false, onesv, (short)0, acc, false, false);
        acc = __builtin_amdgcn_wmma_f32_16x16x4_f32(
                  false, a1, false, onesv, (short)0, acc, false, false);

        // first-occurrence argmax tracking (branchless selects keep EXEC intact)
        bool p;
        p = x.x > bestv; besti = p ? (off + 0) : besti; bestv = p ? x.x : bestv;
        p = x.y > bestv; besti = p ? (off + 1) : besti; bestv = p ? x.y : bestv;
        p = x.z > bestv; besti = p ? (off + 2) : besti; bestv = p ? x.z : bestv;
        p = x.w > bestv; besti = p ? (off + 3) : besti; bestv = p ? x.w : bestv;
    }

    // Collapse WMMA accumulator: all 16 columns identical; lane j holds rows
    // 0..7 (or 8..15 for lane j+16) of column j in acc[0..7].
    float s8 = acc[0] + acc[1] + acc[2] + acc[3] + acc[4] + acc[5] + acc[6] + acc[7];
    s8 += __shfl_xor(s8, 16, 32);          // rows 0..7 + rows 8..15 => wave total

    // Wave-level argmax reduce (max value, min index on ties).
    for (int d = 16; d >= 1; d >>= 1) {
        float ov = __shfl_xor(bestv, d, 32);
        int   oi = __shfl_xor(besti, d, 32);
        if (ov > bestv || (ov == bestv && oi < besti)) { bestv = ov; besti = oi; }
    }

    __shared__ float ssum[8];
    __shared__ float smax[8];
    __shared__ int   sidx[8];
    if (L == 0) { ssum[w] = s8; smax[w] = bestv; sidx[w] = besti; }
    __syncthreads();

    if (tid == 0) {
        float tot = 0.0f;
        float mv = smax[0];
        int   mi = sidx[0];
        for (int i = 0; i < 8; ++i) {
            tot += ssum[i];
            if (smax[i] > mv || (smax[i] == mv && sidx[i] < mi)) {
                mv = smax[i]; mi = sidx[i];
            }
        }
        ws_sum[blk] = tot;
        ws_max[blk] = mv;
        ws_idx[blk] = mi;
    }
}

// ---------------------------------------------------------------------------
// Stage 2: fold over k, compute label loss. One block, 64 threads: one thread
// per (b,s). Total work is tiny (704 * ~30 flops).
// ---------------------------------------------------------------------------
__global__ __launch_bounds__(64) void kploss_stage2(
    const float* __restrict__ cp,
    const float* __restrict__ labels,   // (B, KEYS, 11)
    const float* __restrict__ ws_sum,
    const float* __restrict__ ws_max,
    const int*   __restrict__ ws_idx,
    float* __restrict__ out)            // [64 heat | 64 label]
{
    const int bs = threadIdx.x;         // 0..63 == b*NSTACK + s
    const int b  = bs >> 2;

    float heat = 0.0f;
    float lloss = 0.0f;

    for (int k = 0; k < KEYS; ++k) {
        const int g = bs * KEYS + k;
        heat += ws_sum[g];

        const float* __restrict__ lb = cp + (size_t)(bs * (2 * KEYS) + KEYS + k) * MAP;
        const float* __restrict__ la = labels + (size_t)(b * KEYS + k) * 11;

        const float gx = la[9], gy = la[10];
        const bool valid = (gx > 0.0f) && (gy > 0.0f) &&
                           (gx < (float)HH) && (gy < (float)WW);

        const int idx = ws_idx[g];
        const float xf = (float)(idx / WW);   // index // m
        const float yf = (float)(idx % WW);   // index %  m

        const float dx = gx + la[7] - xf - lb[7];
        const float dy = gy + la[8] - yf - lb[8];
        const float xy = dx * dx + dy * dy;

        const float a = ws_max[g];
        const float conf = (1.0f - a) * (1.0f - a);

        float cls = 0.0f;
        for (int j = 0; j < 7; ++j) {
            const float d = lb[j] - la[j];
            cls += d * d;
        }
        lloss += valid ? (cls + xy + conf) : 0.0f;
    }

    out[bs]      = heat;    // heat_loss (16,4) row-major
    out[64 + bs] = lloss;   // label_loss (16,4) row-major
}

extern "C" void kernel_launch(void* const* d_in, const int* in_sizes, int n_in,
                              void* d_out, int out_size, void* d_ws, size_t ws_size,
                              hipStream_t stream) {
    const float* cp     = (const float*)d_in[0];   // combined_preds
    const float* hmaps  = (const float*)d_in[1];   // heatmaps
    const float* labels = (const float*)d_in[2];   // labels
    float* out = (float*)d_out;

    float* ws_sum = (float*)d_ws;
    float* ws_max = ws_sum + NMAPS;
    int*   ws_idx = (int*)(ws_max + NMAPS);

    kploss_stage1<<<NMAPS, 256, 0, stream>>>(cp, hmaps, ws_sum, ws_max, ws_idx);
    kploss_stage2<<<1, 64, 0, stream>>>(cp, labels, ws_sum, ws_max, ws_idx, out);
}